// Loss_Single_7438883356869
// MI455X (gfx1250) — compile-verified
//
#include <hip/hip_runtime.h>
#include <math.h>

#define NROWS 4096
#define DDIM  128
#define NCLS  10000
#define MARGIN1 0.3f
#define ALPHA 1.0f
#define GAMMA 1.0f

typedef float v2f __attribute__((ext_vector_type(2)));
typedef float v8f __attribute__((ext_vector_type(8)));

// ---------------------------------------------------------------------------
// Kernel 0: per-row squared L2 norm of trip_out
// ---------------------------------------------------------------------------
__global__ __launch_bounds__(128) void row_sq_kernel(const float* __restrict__ X,
                                                     float* __restrict__ sq) {
    __shared__ float red[128];
    const int row = blockIdx.x;
    const float v = X[row * DDIM + threadIdx.x];
    red[threadIdx.x] = v * v;
    __syncthreads();
    for (int s = 64; s > 0; s >>= 1) {
        if (threadIdx.x < s) red[threadIdx.x] += red[threadIdx.x + s];
        __syncthreads();
    }
    if (threadIdx.x == 0) sq[row] = red[0];
}

// ---------------------------------------------------------------------------
// Kernel 1: fused Gram-matrix (fp32 WMMA) -> distances -> hard mining.
// One workgroup per 16-row strip; 8 waves; each wave sweeps 32 column tiles,
// processing TWO tiles concurrently (independent WMMA accumulator chains) to
// hide v_wmma latency. The 4096x4096 distance matrix is never materialized.
// ---------------------------------------------------------------------------
__global__ __launch_bounds__(256) void triplet_kernel(const float* __restrict__ X,
                                                      const float* __restrict__ sq,
                                                      float* __restrict__ trip_row) {
    __shared__ float lsA[16][132];      // 16x128 A strip, padded (132) for bank spread
    __shared__ float red_ap[8][16];
    __shared__ float red_an[8][16];

    const int tid  = threadIdx.x;
    const int wave = tid >> 5;
    const int lane = tid & 31;
    const int half = lane >> 4;         // which K-pair / which 8-row group of acc
    const int l15  = lane & 15;
    const int row0 = blockIdx.x * 16;

    // stage A strip (16 rows x 128) into LDS
    for (int i = tid; i < 16 * DDIM; i += 256) {
        const int r = i >> 7;
        const int c = i & 127;
        lsA[r][c] = X[(row0 + r) * DDIM + c];
    }
    __syncthreads();

    // squared norms for the 8 output rows this lane's accumulator covers
    float sq_r[8];
#pragma unroll
    for (int r = 0; r < 8; ++r) sq_r[r] = sq[row0 + 8 * half + r];

    float ap[8], an[8];
#pragma unroll
    for (int r = 0; r < 8; ++r) { ap[r] = -INFINITY; an[r] = INFINITY; }

    // A fragment base: lane l15 -> matrix row M=l15, k offset 2*half
    const float* aBase = &lsA[l15][2 * half];

    // 32 tiles per wave, processed as 16 pairs with dual accumulators
    for (int j = 0; j < 16; ++j) {
        const int t0 = wave + 8 * (2 * j);
        const int t1 = wave + 8 * (2 * j + 1);
        const int col_0 = t0 * 16 + l15;
        const int col_1 = t1 * 16 + l15;
        const float sq_c0 = sq[col_0];
        const float sq_c1 = sq[col_1];
        const int cid0 = col_0 >> 3;
        const int cid1 = col_1 >> 3;
        const float* bBase0 = &X[col_0 * DDIM + 2 * half];
        const float* bBase1 = &X[col_1 * DDIM + 2 * half];

        v8f acc0 = {0.f, 0.f, 0.f, 0.f, 0.f, 0.f, 0.f, 0.f};
        v8f acc1 = {0.f, 0.f, 0.f, 0.f, 0.f, 0.f, 0.f, 0.f};
#pragma unroll
        for (int k0 = 0; k0 < DDIM; k0 += 4) {
            v2f a  = *reinterpret_cast<const v2f*>(aBase + k0);   // register-resident after hoist
            v2f b0 = *reinterpret_cast<const v2f*>(bBase0 + k0);  // global_load_b64 (L2-hot)
            v2f b1 = *reinterpret_cast<const v2f*>(bBase1 + k0);
            acc0 = __builtin_amdgcn_wmma_f32_16x16x4_f32(
                false, a, false, b0, (short)0, acc0, false, false);
            acc1 = __builtin_amdgcn_wmma_f32_16x16x4_f32(
                false, a, false, b1, (short)0, acc1, false, false);
        }

#pragma unroll
        for (int r = 0; r < 8; ++r) {
            const int row = row0 + 8 * half + r;
            const int rid = row >> 3;

            const float d2_0 = sq_r[r] + sq_c0 - 2.0f * acc0[r];
            const float d0   = sqrtf(fmaxf(d2_0, 1e-12f));
            if (rid == cid0) ap[r] = fmaxf(ap[r], d0);
            else             an[r] = fminf(an[r], d0);

            const float d2_1 = sq_r[r] + sq_c1 - 2.0f * acc1[r];
            const float d1   = sqrtf(fmaxf(d2_1, 1e-12f));
            if (rid == cid1) ap[r] = fmaxf(ap[r], d1);
            else             an[r] = fminf(an[r], d1);
        }
    }

    // reduce over the 16 column-lanes of each half-wave (butterfly)
#pragma unroll
    for (int r = 0; r < 8; ++r) {
#pragma unroll
        for (int m = 1; m < 16; m <<= 1) {
            ap[r] = fmaxf(ap[r], __shfl_xor(ap[r], m, 32));
            an[r] = fminf(an[r], __shfl_xor(an[r], m, 32));
        }
    }
    if (l15 == 0) {
#pragma unroll
        for (int r = 0; r < 8; ++r) {
            red_ap[wave][8 * half + r] = ap[r];
            red_an[wave][8 * half + r] = an[r];
        }
    }
    __syncthreads();

    if (tid < 16) {
        float a_ = -INFINITY, n_ = INFINITY;
#pragma unroll
        for (int w = 0; w < 8; ++w) {
            a_ = fmaxf(a_, red_ap[w][tid]);
            n_ = fminf(n_, red_an[w][tid]);
        }
        trip_row[row0 + tid] = fmaxf(a_ - n_ + MARGIN1, 0.0f);
    }
}

// ---------------------------------------------------------------------------
// Kernel 2: single-pass online-softmax cross-entropy, one block per row.
// float4 loads (global_load_b128) to stream the 164 MB at HBM roofline.
// NCLS = 10000 is divisible by 4 -> exact vectorization.
// ---------------------------------------------------------------------------
__global__ __launch_bounds__(256) void xent_kernel(const float* __restrict__ logits,
                                                   const int* __restrict__ targets,
                                                   float* __restrict__ nll) {
    __shared__ float sm[256];
    __shared__ float ss[256];
    const int row = blockIdx.x;
    const int tid = threadIdx.x;
    const float* rowp = logits + (size_t)row * NCLS;
    const float4* rowp4 = reinterpret_cast<const float4*>(rowp);

    float m = -INFINITY, s = 0.0f;
    for (int c = tid; c < NCLS / 4; c += 256) {
        const float4 x = rowp4[c];
        const float mx = fmaxf(fmaxf(x.x, x.y), fmaxf(x.z, x.w));
        const float nm = fmaxf(m, mx);
        s = s * __expf(m - nm)
          + __expf(x.x - nm) + __expf(x.y - nm)
          + __expf(x.z - nm) + __expf(x.w - nm);
        m = nm;
    }
    sm[tid] = m; ss[tid] = s;
    __syncthreads();
    for (int st = 128; st > 0; st >>= 1) {
        if (tid < st) {
            const float m2 = sm[tid + st], s2 = ss[tid + st];
            const float M  = fmaxf(sm[tid], m2);
            ss[tid] = ss[tid] * __expf(sm[tid] - M) + s2 * __expf(m2 - M);
            sm[tid] = M;
        }
        __syncthreads();
    }
    if (tid == 0) {
        const float lse = sm[0] + __logf(ss[0]);
        nll[row] = lse - rowp[targets[row]];
    }
}

// ---------------------------------------------------------------------------
// Kernel 3: final scalar: ALPHA*mean(trip) + GAMMA*mean(nll)
// ---------------------------------------------------------------------------
__global__ __launch_bounds__(256) void finalize_kernel(const float* __restrict__ trip_row,
                                                       const float* __restrict__ nll,
                                                       float* __restrict__ out) {
    __shared__ float red[256];
    float acc = 0.0f;
    for (int i = threadIdx.x; i < NROWS; i += 256)
        acc += ALPHA * trip_row[i] + GAMMA * nll[i];
    red[threadIdx.x] = acc;
    __syncthreads();
    for (int s = 128; s > 0; s >>= 1) {
        if (threadIdx.x < s) red[threadIdx.x] += red[threadIdx.x + s];
        __syncthreads();
    }
    if (threadIdx.x == 0) out[0] = red[0] / (float)NROWS;
}

// ---------------------------------------------------------------------------
extern "C" void kernel_launch(void* const* d_in, const int* in_sizes, int n_in,
                              void* d_out, int out_size, void* d_ws, size_t ws_size,
                              hipStream_t stream) {
    const float* softmax_out = (const float*)d_in[0];   // [N, C] fp32
    const float* trip_out    = (const float*)d_in[1];   // [N, D] fp32
    const int*   targets     = (const int*)d_in[2];     // [N]    int32
    float* out = (float*)d_out;

    float* sq       = (float*)d_ws;        // [N]
    float* trip_row = sq + NROWS;          // [N]
    float* nll      = trip_row + NROWS;    // [N]

    row_sq_kernel<<<NROWS, 128, 0, stream>>>(trip_out, sq);
    xent_kernel<<<NROWS, 256, 0, stream>>>(softmax_out, targets, nll);
    triplet_kernel<<<NROWS / 16, 256, 0, stream>>>(trip_out, sq, trip_row);
    finalize_kernel<<<1, 256, 0, stream>>>(trip_row, nll, out);
}